// SocialRecommender_87866440942245
// MI455X (gfx1250) — compile-verified
//
#include <hip/hip_runtime.h>
#include <math.h>

typedef float v2f __attribute__((ext_vector_type(2)));
typedef float v8f __attribute__((ext_vector_type(8)));

#define D_DIM 64
#define B_SZ  2048
#define LEAK_F 0.2f
#define INV_TAU 5.0f   // 1/0.2

// ---------------------------------------------------------------------------
// SpMM: out[row] += val * x[col], 64-wide rows, 4 edges per 256-thread block.
// Coalesced along D; native f32 global atomics for the scatter.
// ---------------------------------------------------------------------------
__global__ void spmm_atomic_kernel(const int* __restrict__ rows,
                                   const int* __restrict__ cols,
                                   const float* __restrict__ vals,
                                   const float* __restrict__ x,
                                   float* __restrict__ out, int nnz) {
  int e = blockIdx.x * 4 + (threadIdx.x >> 6);
  int d = threadIdx.x & 63;
  if (e >= nnz) return;
  int r = rows[e];
  int c = cols[e];
  float v = vals[e];
  atomicAdd(out + (size_t)r * D_DIM + d, v * x[(size_t)c * D_DIM + d]);
}

// ---------------------------------------------------------------------------
// out[row] = LayerNorm(base[row] + t[row]). One wave32 per row, float2/lane.
// ---------------------------------------------------------------------------
__global__ void add_layernorm_kernel(const float* __restrict__ base,
                                     const float* __restrict__ t,
                                     const float* __restrict__ gamma,
                                     const float* __restrict__ beta,
                                     float* __restrict__ out, int n) {
  int row = blockIdx.x * 8 + (threadIdx.x >> 5);
  int ln  = threadIdx.x & 31;
  if (row >= n) return;
  float2 xb = ((const float2*)(base + (size_t)row * D_DIM))[ln];
  float2 xt = ((const float2*)(t    + (size_t)row * D_DIM))[ln];
  float x0 = xb.x + xt.x, x1 = xb.y + xt.y;
  float s = x0 + x1;
  for (int o = 16; o > 0; o >>= 1) s += __shfl_xor(s, o, 32);
  float mu = s * (1.0f / 64.0f);
  float c0 = x0 - mu, c1 = x1 - mu;
  float v = c0 * c0 + c1 * c1;
  for (int o = 16; o > 0; o >>= 1) v += __shfl_xor(v, o, 32);
  float rs = rsqrtf(v * (1.0f / 64.0f) + 1e-5f);
  float2 g = ((const float2*)gamma)[ln];
  float2 b = ((const float2*)beta)[ln];
  float2 o2;
  o2.x = g.x * c0 * rs + b.x;
  o2.y = g.y * c1 * rs + b.y;
  ((float2*)(out + (size_t)row * D_DIM))[ln] = o2;
}

// ---------------------------------------------------------------------------
// Gathered LayerNorm: out[b] = LN(base[idx[b]] + t[idx[b]]), compact [B,64].
// Layer-2 embeddings are only ever consumed at batch rows -> skip full pass.
// ---------------------------------------------------------------------------
__global__ void add_layernorm_gather_kernel(const float* __restrict__ base,
                                            const float* __restrict__ t,
                                            const float* __restrict__ gamma,
                                            const float* __restrict__ beta,
                                            const int* __restrict__ idx,
                                            float* __restrict__ out, int B) {
  int b  = blockIdx.x * 8 + (threadIdx.x >> 5);
  int ln = threadIdx.x & 31;
  if (b >= B) return;
  int row = idx[b];
  float2 xb = ((const float2*)(base + (size_t)row * D_DIM))[ln];
  float2 xt = ((const float2*)(t    + (size_t)row * D_DIM))[ln];
  float x0 = xb.x + xt.x, x1 = xb.y + xt.y;
  float s = x0 + x1;
  for (int o = 16; o > 0; o >>= 1) s += __shfl_xor(s, o, 32);
  float mu = s * (1.0f / 64.0f);
  float c0 = x0 - mu, c1 = x1 - mu;
  float v = c0 * c0 + c1 * c1;
  for (int o = 16; o > 0; o >>= 1) v += __shfl_xor(v, o, 32);
  float rs = rsqrtf(v * (1.0f / 64.0f) + 1e-5f);
  float2 g = ((const float2*)gamma)[ln];
  float2 bb = ((const float2*)beta)[ln];
  float2 o2;
  o2.x = g.x * c0 * rs + bb.x;
  o2.y = g.y * c1 * rs + bb.y;
  ((float2*)(out + (size_t)b * D_DIM))[ln] = o2;
}

// ---------------------------------------------------------------------------
// Per-batch-row attention over 3 layer embeddings -> cat[b, off..off+63].
// e0/e1 indexed by users[b] (full tables); e2 compact [B,64] indexed by b.
// ---------------------------------------------------------------------------
__global__ void attn_cat_kernel(const float* __restrict__ e0,
                                const float* __restrict__ e1,
                                const float* __restrict__ e2b,
                                const float* __restrict__ w,
                                const int* __restrict__ users,
                                float* __restrict__ cat, int off, int B) {
  int b  = blockIdx.x * 8 + (threadIdx.x >> 5);
  int ln = threadIdx.x & 31;
  if (b >= B) return;
  int u = users[b];
  float2 v0 = ((const float2*)(e0  + (size_t)u * D_DIM))[ln];
  float2 v1 = ((const float2*)(e1  + (size_t)u * D_DIM))[ln];
  float2 v2 = ((const float2*)(e2b + (size_t)b * D_DIM))[ln];
  float2 ww = ((const float2*)w)[ln];
  float s0 = v0.x * ww.x + v0.y * ww.y;
  float s1 = v1.x * ww.x + v1.y * ww.y;
  float s2 = v2.x * ww.x + v2.y * ww.y;
  for (int o = 16; o > 0; o >>= 1) {
    s0 += __shfl_xor(s0, o, 32);
    s1 += __shfl_xor(s1, o, 32);
    s2 += __shfl_xor(s2, o, 32);
  }
  float mx = fmaxf(s0, fmaxf(s1, s2));
  float a0 = __expf(s0 - mx), a1 = __expf(s1 - mx), a2 = __expf(s2 - mx);
  float inv = 1.0f / (a0 + a1 + a2);
  a0 *= inv; a1 *= inv; a2 *= inv;
  float2 o2;
  o2.x = a0 * v0.x + a1 * v1.x + a2 * v2.x;
  o2.y = a0 * v0.y + a1 * v1.y + a2 * v2.y;
  ((float2*)(cat + (size_t)b * 192 + off))[ln] = o2;
}

// Raw gather of user rows into cat[:, off..off+63].
__global__ void gather_cat_kernel(const float* __restrict__ src,
                                  const int* __restrict__ idx,
                                  float* __restrict__ cat, int off, int B) {
  int b = blockIdx.x * 4 + (threadIdx.x >> 6);
  int d = threadIdx.x & 63;
  if (b >= B) return;
  cat[(size_t)b * 192 + off + d] = src[(size_t)idx[b] * D_DIM + d];
}

// ---------------------------------------------------------------------------
// f32 WMMA GEMM: C[M,N] = act(A[M,K] @ op(B) + bias)
//   BT==0: B is [K,N] row-major; BT==1: B is [N,K] row-major -> A @ B^T.
// One wave32 per 16x16 C tile; V_WMMA_F32_16X16X4_F32, K stepped by 4.
// A-frag (16x4 f32): lanes 0-15 hold K={k,k+1}, lanes 16-31 hold K={k+2,k+3}.
// C-frag: VGPR r -> row tileM + r + 8*half, col tileN + (lane&15).
// ---------------------------------------------------------------------------
template<int BT, int LEAKY>
__global__ void wmma_gemm_f32_kernel(const float* __restrict__ A,
                                     const float* __restrict__ B,
                                     const float* __restrict__ bias,
                                     float* __restrict__ C,
                                     int M, int N, int K,
                                     int lda, int ldb, int ldc) {
  int tilesN = N >> 4;
  int tileN = (blockIdx.x % tilesN) << 4;
  int tileM = (blockIdx.x / tilesN) << 4;
  int lane = threadIdx.x & 31;
  int half = lane >> 4;      // 0: K pair {0,1}, 1: K pair {2,3}
  int l    = lane & 15;

  v8f acc = {};
  for (int k = 0; k < K; k += 4) {
    v2f a, b;
    const float* ap = A + (size_t)(tileM + l) * lda + (k + 2 * half);
    a.x = ap[0];
    a.y = ap[1];
    if (BT) {
      const float* bp = B + (size_t)(tileN + l) * ldb + (k + 2 * half);
      b.x = bp[0];
      b.y = bp[1];
    } else {
      const float* bp = B + (size_t)(k + 2 * half) * ldb + (tileN + l);
      b.x = bp[0];
      b.y = bp[ldb];
    }
    acc = __builtin_amdgcn_wmma_f32_16x16x4_f32(
        /*neg_a=*/false, a, /*neg_b=*/false, b,
        /*c_mod=*/(short)0, acc, /*reuse_a=*/false, /*reuse_b=*/false);
  }

#pragma unroll
  for (int r = 0; r < 8; ++r) {
    int row = tileM + r + 8 * half;
    int col = tileN + l;
    float v = acc[r];
    if (bias) v += bias[col];
    if (LEAKY) v = (v >= 0.0f) ? v : LEAK_F * v;
    C[(size_t)row * ldc + col] = v;
  }
}

// ---------------------------------------------------------------------------
// In-place row softmax of (sim + gumbel) * (1/TAU). One 256-thread block per
// row; each thread reads exactly the 8 elements it later writes (safe in place).
// ---------------------------------------------------------------------------
__global__ void gumbel_softmax_kernel(const float* __restrict__ gum,
                                      float* __restrict__ io) {
  int row = blockIdx.x;
  int tid = threadIdx.x;
  int w = tid >> 5, ln = tid & 31;
  float* sp = io + (size_t)row * B_SZ;
  const float* gp = gum + (size_t)row * B_SZ;
  float z[8];
  float m = -1e30f;
#pragma unroll
  for (int i = 0; i < 8; ++i) {
    int c = tid + i * 256;
    z[i] = (sp[c] + gp[c]) * INV_TAU;
    m = fmaxf(m, z[i]);
  }
  for (int o = 16; o > 0; o >>= 1) m = fmaxf(m, __shfl_xor(m, o, 32));
  __shared__ float red[8];
  if (ln == 0) red[w] = m;
  __syncthreads();
  if (w == 0) {
    float t = (ln < 8) ? red[ln] : -1e30f;
    for (int o = 16; o > 0; o >>= 1) t = fmaxf(t, __shfl_xor(t, o, 32));
    if (ln == 0) red[0] = t;
  }
  __syncthreads();
  m = red[0];
  __syncthreads();
  float s = 0.0f;
#pragma unroll
  for (int i = 0; i < 8; ++i) { z[i] = __expf(z[i] - m); s += z[i]; }
  for (int o = 16; o > 0; o >>= 1) s += __shfl_xor(s, o, 32);
  if (ln == 0) red[w] = s;
  __syncthreads();
  if (w == 0) {
    float t = (ln < 8) ? red[ln] : 0.0f;
    for (int o = 16; o > 0; o >>= 1) t += __shfl_xor(t, o, 32);
    if (ln == 0) red[0] = t;
  }
  __syncthreads();
  float inv = 1.0f / red[0];
#pragma unroll
  for (int i = 0; i < 8; ++i) sp[tid + i * 256] = z[i] * inv;
}

// ---------------------------------------------------------------------------
// Fused final-item LayerNorm + scoring:
//   i2[row] = LN(i1[row] + it[row]);  score = dot(uf[b], i2[item])
// Avoids materializing the full [I,64] layer-2 item embedding.
// ---------------------------------------------------------------------------
__device__ __forceinline__ float ln_dot(const float* __restrict__ i1,
                                        const float* __restrict__ it,
                                        float2 g, float2 bb,
                                        int item, int ln, float2 uv) {
  float2 xb = ((const float2*)(i1 + (size_t)item * D_DIM))[ln];
  float2 xt = ((const float2*)(it + (size_t)item * D_DIM))[ln];
  float x0 = xb.x + xt.x, x1 = xb.y + xt.y;
  float s = x0 + x1;
  for (int o = 16; o > 0; o >>= 1) s += __shfl_xor(s, o, 32);
  float mu = s * (1.0f / 64.0f);
  float c0 = x0 - mu, c1 = x1 - mu;
  float v = c0 * c0 + c1 * c1;
  for (int o = 16; o > 0; o >>= 1) v += __shfl_xor(v, o, 32);
  float rs = rsqrtf(v * (1.0f / 64.0f) + 1e-5f);
  float y0 = g.x * c0 * rs + bb.x;
  float y1 = g.y * c1 * rs + bb.y;
  float d = uv.x * y0 + uv.y * y1;
  for (int o = 16; o > 0; o >>= 1) d += __shfl_xor(d, o, 32);
  return d;
}

__global__ void scores_ln_kernel(const float* __restrict__ uf,
                                 const float* __restrict__ i1,
                                 const float* __restrict__ it,
                                 const float* __restrict__ gamma,
                                 const float* __restrict__ beta,
                                 const int* __restrict__ pos,
                                 const int* __restrict__ neg,
                                 float* __restrict__ out, int B) {
  int b  = blockIdx.x * 8 + (threadIdx.x >> 5);
  int ln = threadIdx.x & 31;
  if (b >= B) return;
  float2 uv = ((const float2*)(uf + (size_t)b * D_DIM))[ln];
  float2 g  = ((const float2*)gamma)[ln];
  float2 bb = ((const float2*)beta)[ln];
  float ps = ln_dot(i1, it, g, bb, pos[b], ln, uv);
  float ns = ln_dot(i1, it, g, bb, neg[b], ln, uv);
  if (ln == 0) {
    out[(size_t)b * 2 + 0] = ps;
    out[(size_t)b * 2 + 1] = ns;
  }
}

// ---------------------------------------------------------------------------
extern "C" void kernel_launch(void* const* d_in, const int* in_sizes, int n_in,
                              void* d_out, int out_size, void* d_ws, size_t ws_size,
                              hipStream_t stream) {
  const int U = 200000, I = 100000, B = B_SZ;
  const int NNZ_UI = 2000000, NNZ_S = 1000000;
  const size_t UD = (size_t)U * D_DIM, ID = (size_t)I * D_DIM, BD = (size_t)B * D_DIM;

  // Inputs (setup_inputs dict order)
  const float* user_emb  = (const float*)d_in[0];
  const float* item_emb  = (const float*)d_in[1];
  const float* ln_gamma  = (const float*)d_in[2];
  const float* ln_beta   = (const float*)d_in[3];
  const float* ui_attn_w = (const float*)d_in[4];
  const float* so_attn_w = (const float*)d_in[5];
  const float* samp_W    = (const float*)d_in[6];
  const float* out_W1    = (const float*)d_in[7];   // [64,192]
  const float* out_b1    = (const float*)d_in[8];
  const float* out_W2    = (const float*)d_in[9];   // [64,64]
  const float* out_b2    = (const float*)d_in[10];
  const float* ui_vals   = (const float*)d_in[11];
  const float* so_vals   = (const float*)d_in[12];
  const int*   users     = (const int*)d_in[13];
  const int*   pos_items = (const int*)d_in[14];
  const int*   neg_items = (const int*)d_in[15];
  const int*   ui_rows   = (const int*)d_in[16];
  const int*   ui_cols   = (const int*)d_in[17];
  const int*   so_rows   = (const int*)d_in[18];
  const int*   so_cols   = (const int*)d_in[19];
  const float* gumbel    = (const float*)d_in[20];

  // Workspace layout (floats) — ~210 MB total
  float* f    = (float*)d_ws;
  float* u1   = f;                    // U x 64   (ui layer-1 user emb, kept)
  float* s1   = u1 + UD;              // U x 64   (social layer-1, kept)
  float* utmp = s1 + UD;              // U x 64   (shared SpMM accumulator)
  float* i1   = utmp + UD;            // I x 64   (ui layer-1 item emb, kept)
  float* itmp = i1 + ID;              // I x 64   (item SpMM accumulator; layer-2
                                      //           residual kept live for scores)
  float* u2b  = itmp + ID;            // B x 64   (layer-2 user emb @ batch rows)
  float* s2b  = u2b + BD;             // B x 64   (layer-2 social emb @ batch rows)
  float* cat  = s2b + BD;             // B x 192  ([ui | soc | be])
  float* beW  = cat + (size_t)B * 192;// B x 64
  float* h    = beW + BD;             // B x 64
  float* uf   = h + BD;               // B x 64

  float* out_scores  = (float*)d_out;                   // B x 2
  float* out_softmax = (float*)d_out + (size_t)B * 2;   // B x B (sim in place)

  dim3 blk256(256);
  int gUI   = (NNZ_UI + 3) / 4;
  int gS    = (NNZ_S + 3) / 4;
  int gLN_U = (U + 7) / 8;
  int gLN_I = (I + 7) / 8;
  int gB8   = (B + 7) / 8;
  int gB4   = (B + 3) / 4;

  // ---- GraphConv layer 1 (ui bipartite) ----
  hipMemsetAsync(utmp, 0, UD * sizeof(float), stream);
  hipMemsetAsync(itmp, 0, ID * sizeof(float), stream);
  spmm_atomic_kernel<<<gUI, blk256, 0, stream>>>(ui_rows, ui_cols, ui_vals, item_emb, utmp, NNZ_UI);
  spmm_atomic_kernel<<<gUI, blk256, 0, stream>>>(ui_cols, ui_rows, ui_vals, user_emb, itmp, NNZ_UI);
  add_layernorm_kernel<<<gLN_U, blk256, 0, stream>>>(user_emb, utmp, ln_gamma, ln_beta, u1, U);
  add_layernorm_kernel<<<gLN_I, blk256, 0, stream>>>(item_emb, itmp, ln_gamma, ln_beta, i1, I);

  // ---- GraphConv layer 2: full SpMM, but LN only where consumed ----
  hipMemsetAsync(utmp, 0, UD * sizeof(float), stream);
  hipMemsetAsync(itmp, 0, ID * sizeof(float), stream);
  spmm_atomic_kernel<<<gUI, blk256, 0, stream>>>(ui_rows, ui_cols, ui_vals, i1, utmp, NNZ_UI);
  spmm_atomic_kernel<<<gUI, blk256, 0, stream>>>(ui_cols, ui_rows, ui_vals, u1, itmp, NNZ_UI);
  add_layernorm_gather_kernel<<<gB8, blk256, 0, stream>>>(u1, utmp, ln_gamma, ln_beta, users, u2b, B);
  // itmp (= layer-2 item residual) stays live; final item LN fused into scores.

  // ---- Social layers (utmp reused after u2b consumed it) ----
  hipMemsetAsync(utmp, 0, UD * sizeof(float), stream);
  spmm_atomic_kernel<<<gS, blk256, 0, stream>>>(so_rows, so_cols, so_vals, user_emb, utmp, NNZ_S);
  add_layernorm_kernel<<<gLN_U, blk256, 0, stream>>>(user_emb, utmp, ln_gamma, ln_beta, s1, U);
  hipMemsetAsync(utmp, 0, UD * sizeof(float), stream);
  spmm_atomic_kernel<<<gS, blk256, 0, stream>>>(so_rows, so_cols, so_vals, s1, utmp, NNZ_S);
  add_layernorm_gather_kernel<<<gB8, blk256, 0, stream>>>(s1, utmp, ln_gamma, ln_beta, users, s2b, B);

  // ---- Attention aggregation at batch rows + concat build ----
  attn_cat_kernel<<<gB8, blk256, 0, stream>>>(user_emb, u1, u2b, ui_attn_w, users, cat, 0, B);
  attn_cat_kernel<<<gB8, blk256, 0, stream>>>(user_emb, s1, s2b, so_attn_w, users, cat, 64, B);
  gather_cat_kernel<<<gB4, blk256, 0, stream>>>(user_emb, users, cat, 128, B);

  const float* be = cat + 128;  // [B,64] with row stride 192

  // ---- DualSampling: sim = (be @ W) @ be^T directly into d_out, softmax in place
  wmma_gemm_f32_kernel<0, 0><<<(B / 16) * (64 / 16), 32, 0, stream>>>(
      be, samp_W, nullptr, beW, B, 64, 64, 192, 64, 64);
  wmma_gemm_f32_kernel<1, 0><<<(B / 16) * (B / 16), 32, 0, stream>>>(
      beW, be, nullptr, out_softmax, B, B, 64, 64, 192, B);
  gumbel_softmax_kernel<<<B, blk256, 0, stream>>>(gumbel, out_softmax);

  // ---- Output MLP ----
  wmma_gemm_f32_kernel<1, 1><<<(B / 16) * (64 / 16), 32, 0, stream>>>(
      cat, out_W1, out_b1, h, B, 64, 192, 192, 192, 64);
  wmma_gemm_f32_kernel<1, 0><<<(B / 16) * (64 / 16), 32, 0, stream>>>(
      h, out_W2, out_b2, uf, B, 64, 64, 64, 64, 64);

  // ---- Scores with fused final-item LayerNorm ----
  scores_ln_kernel<<<gB8, blk256, 0, stream>>>(uf, i1, itmp, ln_gamma, ln_beta,
                                               pos_items, neg_items, out_scores, B);
}